// RetinaNet_31336081392206
// MI455X (gfx1250) — compile-verified
//
#include <hip/hip_runtime.h>
#include <hip/hip_bf16.h>
#include <stdint.h>

// ---------------- problem constants (match reference) ----------------
#define B_        2
#define N_        49104            // divisible by 4
#define NV_       (N_ / 4)         // 12276 float4 chunks
#define C_        20
#define MAXDET    300
#define IOU_THR_  0.5f
#define SCORE_THR_ 0.05f
#define STD_      0.2f
#define NEG_      (-1.0e9f)
#define VALID_CUT (-5.0e8f)        // NEG * 0.5
#define HI_XY     511.0f           // W-1 == H-1

// ---------------- TDM (Tensor Data Mover) path ----------------
#if defined(__has_builtin)
#if __has_builtin(__builtin_amdgcn_tensor_load_to_lds)
#define USE_TDM 1
#endif
#endif
#ifndef USE_TDM
#define USE_TDM 0
#endif

#if USE_TDM
typedef unsigned int tdm_u32x4 __attribute__((ext_vector_type(4)));
typedef int          tdm_i32x8 __attribute__((ext_vector_type(8)));
typedef int          tdm_i32x4 __attribute__((ext_vector_type(4)));

// One-shot 1D DMA: nelem f32 elements, global -> LDS. nelem must be <= 65535.
__device__ __forceinline__ void tdm_load_1d_f32(uint32_t lds_off,
                                                const void* gptr,
                                                uint32_t nelem) {
  uint64_t ga = (uint64_t)(uintptr_t)gptr;
  tdm_u32x4 g0;
  g0[0] = 1u;                                            // count=1, user mode
  g0[1] = lds_off;                                       // lds_addr (bytes)
  g0[2] = (uint32_t)ga;                                  // global_addr[31:0]
  g0[3] = (uint32_t)((ga >> 32) & 0x01FFFFFFu)           // global_addr[56:32]
        | (2u << 30);                                    // type=2 ("image")
  tdm_i32x8 g1;
  g1[0] = (int)(2u << 16);                               // data_size=4B, no mask
  g1[1] = (int)((nelem & 0xFFFFu) << 16);                // tensor_dim0[15:0]
  g1[2] = (int)(((nelem >> 16) & 0xFFFFu) | (1u << 16)); // tensor_dim0[31:16], tensor_dim1=1
  g1[3] = (int)((nelem & 0xFFFFu) << 16);                // tile_dim0 = nelem
  g1[4] = 0;                                             // tile_dim1/2 unused
  g1[5] = (int)nelem;                                    // tensor_dim0_stride (low)
  g1[6] = 0;
  g1[7] = 0;
  tdm_i32x4 z4 = {0, 0, 0, 0};                           // groups 2/3 disabled (<=2D)
  tdm_i32x8 z8 = {0, 0, 0, 0, 0, 0, 0, 0};
  __builtin_amdgcn_tensor_load_to_lds(g0, g1, z4, z4, z8, 0);
}
#endif

// ---------------- kernel 1: decode + clip + transpose/threshold scores -----
__global__ __launch_bounds__(256) void rn_decode_kernel(
    const float* __restrict__ anchors,   // [B,N,4]
    const float* __restrict__ deltas,    // [B,N,4]
    const float* __restrict__ cls,       // [B,N,C]
    float* __restrict__ boxes,           // [B,N,4]  (ws)
    float* __restrict__ scoresT) {       // [B,C,N]  (ws)
  int g = blockIdx.x * blockDim.x + threadIdx.x;
  if (g >= B_ * N_) return;
  int b = g / N_;
  int n = g - b * N_;

  const float4 a = reinterpret_cast<const float4*>(anchors)[g];
  const float4 d = reinterpret_cast<const float4*>(deltas)[g];
  float w = a.z - a.x, h = a.w - a.y;
  float x1 = a.x + d.x * (STD_ * w);
  float y1 = a.y + d.y * (STD_ * h);
  float x2 = a.z + d.z * (STD_ * w);
  float y2 = a.w + d.w * (STD_ * h);
  x1 = fminf(fmaxf(x1, 0.0f), HI_XY);
  y1 = fminf(fmaxf(y1, 0.0f), HI_XY);
  x2 = fminf(fmaxf(x2, 0.0f), HI_XY);
  y2 = fminf(fmaxf(y2, 0.0f), HI_XY);
  reinterpret_cast<float4*>(boxes)[g] = make_float4(x1, y1, x2, y2);

  const float* c = cls + (size_t)g * C_;
  float* st = scoresT + (size_t)b * C_ * N_ + n;
#pragma unroll
  for (int cc = 0; cc < C_; ++cc) {
    float s = c[cc];
    st[(size_t)cc * N_] = (s > SCORE_THR_) ? s : NEG_;   // coalesced across n
  }
}

// ---------------- kernel 2: per-(b,c) greedy NMS -------------------------
// Scores live in LDS (196 KB, possible only with CDNA5's 320KB WGP LDS).
// Each thread exclusively owns interleaved float4 chunks of the score array,
// so score updates need NO barriers; suppression is fused with computing the
// next iteration's per-thread argmax (one LDS sweep per NMS step, ds_*_b128).
__global__ __launch_bounds__(1024) void rn_nms_kernel(
    const float* __restrict__ boxes,     // [B,N,4]
    const float* __restrict__ scoresT,   // [B,C,N]
    float* __restrict__ out_sc,          // [B*C, MAXDET]
    int*   __restrict__ out_idx) {       // [B*C, MAXDET]
  extern __shared__ float sS[];          // N_ floats = 196416 B
  __shared__ float sWMax[32];
  __shared__ int   sWIdx[32];
  __shared__ float sBox[4];
  __shared__ float sBS;
  __shared__ int   sBI;

  const int bc  = blockIdx.x;            // 0 .. B*C-1
  const int b   = bc / C_;
  const int tid = threadIdx.x;
  const int BT  = blockDim.x;            // 1024 = 32 wave32 waves
  const float* src = scoresT + (size_t)bc * N_;
  const float4* bx4 = reinterpret_cast<const float4*>(boxes + (size_t)b * N_ * 4);
  float4* sS4 = reinterpret_cast<float4*>(sS);

  // ---- bulk-load the score row into LDS via the Tensor Data Mover ----
#if USE_TDM
  if ((tid >> 5) == 0) {                 // one wave issues the DMA (EXEC ignored)
    tdm_load_1d_f32((uint32_t)(uintptr_t)sS, src, (uint32_t)N_);
    __builtin_amdgcn_s_wait_tensorcnt(0);
  }
#else
  for (int i = tid; i < N_; i += BT) sS[i] = src[i];
#endif
  __syncthreads();

  float* osc = out_sc + (size_t)bc * MAXDET;
  int*   oix = out_idx + (size_t)bc * MAXDET;

  // ---- initial per-thread argmax over this thread's chunks ----
  float best = -3.4e38f;
  int   bi   = 0;
  for (int v = tid; v < NV_; v += BT) {
    const float4 sv = sS4[v];
    const int base = v * 4;
    if (sv.x > best) { best = sv.x; bi = base; }
    if (sv.y > best) { best = sv.y; bi = base + 1; }
    if (sv.z > best) { best = sv.z; bi = base + 2; }
    if (sv.w > best) { best = sv.w; bi = base + 3; }
  }

  for (int it = 0; it < MAXDET; ++it) {
    // ---- block argmax reduction (lowest-index tie-break) ----
    float rb = best;
    int   ri = bi;
#pragma unroll
    for (int off = 16; off > 0; off >>= 1) {
      float os = __shfl_down(rb, off, 32);
      int   oi = __shfl_down(ri, off, 32);
      if (os > rb || (os == rb && oi < ri)) { rb = os; ri = oi; }
    }
    if ((tid & 31) == 0) { sWMax[tid >> 5] = rb; sWIdx[tid >> 5] = ri; }
    __syncthreads();
    if (tid < 32) {
      float b2 = sWMax[tid];
      int   i2 = sWIdx[tid];
#pragma unroll
      for (int off = 16; off > 0; off >>= 1) {
        float os = __shfl_down(b2, off, 32);
        int   oi = __shfl_down(i2, off, 32);
        if (os > b2 || (os == b2 && oi < i2)) { b2 = os; i2 = oi; }
      }
      if (tid == 0) {
        sBS = b2;
        sBI = i2;
        const float4 bb = bx4[i2];
        sBox[0] = bb.x; sBox[1] = bb.y; sBox[2] = bb.z; sBox[3] = bb.w;
      }
    }
    __syncthreads();

    const float bs = sBS;

    if (bs <= VALID_CUT) {
      // All remaining candidates suppressed: the fixed-length reference scan
      // emits (argmax==0, NEG) for every remaining step.
      for (int k = it + tid; k < MAXDET; k += BT) { osc[k] = NEG_; oix[k] = 0; }
      break;                              // uniform across block
    }
    if (tid == 0) { osc[it] = bs; oix[it] = sBI; }

    // ---- fused: IoU suppression + next-iteration local argmax ----
    const float bx1 = sBox[0], by1 = sBox[1], bx2 = sBox[2], by2 = sBox[3];
    const float ab  = (bx2 - bx1) * (by2 - by1);
    best = -3.4e38f;
    bi   = 0;
    for (int v = tid; v < NV_; v += BT) {
      float4 sv = sS4[v];
      bool changed = false;
      const int base = v * 4;
#define RN_PROC(COMP, J)                                                      \
      { float s = sv.COMP;                                                    \
        if (s > VALID_CUT) {                                                  \
          const float4 o = bx4[base + (J)];                                   \
          float xx1 = fmaxf(bx1, o.x), yy1 = fmaxf(by1, o.y);                 \
          float xx2 = fminf(bx2, o.z), yy2 = fminf(by2, o.w);                 \
          float inter = fmaxf(xx2 - xx1, 0.0f) * fmaxf(yy2 - yy1, 0.0f);      \
          float oarea = (o.z - o.x) * (o.w - o.y);                            \
          float iou = inter / (ab + oarea - inter + 1e-8f);                   \
          if (iou > IOU_THR_) { sv.COMP = NEG_; changed = true; }             \
          else if (s > best) { best = s; bi = base + (J); }                   \
        } }
      RN_PROC(x, 0)
      RN_PROC(y, 1)
      RN_PROC(z, 2)
      RN_PROC(w, 3)
#undef RN_PROC
      if (changed) sS4[v] = sv;           // thread-private chunk: no barrier
    }
    // next loop iteration's first barrier orders sWMax/sBox reuse
  }
}

// ---------------- kernel 3: per-batch global top-300 + gather --------------
__global__ __launch_bounds__(256) void rn_topk_kernel(
    const float* __restrict__ boxes,     // [B,N,4]
    const float* __restrict__ nsc,       // [B, C*MAXDET]
    const int*   __restrict__ nidx,      // [B, C*MAXDET]
    float* __restrict__ out) {           // boxes[B,300,4] | scores[B,300] | labels(i32)[B,300]
  const int b = blockIdx.x;
  const int M = C_ * MAXDET;             // 6000
  __shared__ float ssc[C_ * MAXDET];
  __shared__ int   sidx[C_ * MAXDET];
  __shared__ float rM[8];
  __shared__ int   rI[8];
  const int tid = threadIdx.x;

  for (int i = tid; i < M; i += 256) {
    ssc[i]  = nsc[(size_t)b * M + i];
    sidx[i] = nidx[(size_t)b * M + i];
  }
  __syncthreads();

  float* ob  = out + (size_t)b * MAXDET * 4;
  float* osx = out + (size_t)B_ * MAXDET * 4 + (size_t)b * MAXDET;
  int*   ol  = reinterpret_cast<int*>(out + (size_t)B_ * MAXDET * 4 +
                                      (size_t)B_ * MAXDET) + (size_t)b * MAXDET;
  const float* bx = boxes + (size_t)b * N_ * 4;

  for (int k = 0; k < MAXDET; ++k) {
    float best = -3.4e38f;
    int   bi   = 0;
    for (int i = tid; i < M; i += 256) {
      float s = ssc[i];
      if (s > best) { best = s; bi = i; }
    }
#pragma unroll
    for (int off = 16; off > 0; off >>= 1) {
      float os = __shfl_down(best, off, 32);
      int   oi = __shfl_down(bi, off, 32);
      if (os > best || (os == best && oi < bi)) { best = os; bi = oi; }
    }
    if ((tid & 31) == 0) { rM[tid >> 5] = best; rI[tid >> 5] = bi; }
    __syncthreads();
    if (tid == 0) {
#pragma unroll
      for (int wv = 1; wv < 8; ++wv)
        if (rM[wv] > best || (rM[wv] == best && rI[wv] < bi)) { best = rM[wv]; bi = rI[wv]; }
      ssc[bi] = -3.4e38f;                 // remove from further selection
      const bool valid = best > VALID_CUT;
      osx[k] = valid ? best : -1.0f;
      ol[k]  = valid ? (bi / MAXDET) : -1;
      if (valid) {
        const float4 bb = reinterpret_cast<const float4*>(bx)[sidx[bi]];
        ob[4 * k + 0] = bb.x; ob[4 * k + 1] = bb.y;
        ob[4 * k + 2] = bb.z; ob[4 * k + 3] = bb.w;
      } else {
        ob[4 * k + 0] = -1.0f; ob[4 * k + 1] = -1.0f;
        ob[4 * k + 2] = -1.0f; ob[4 * k + 3] = -1.0f;
      }
    }
    __syncthreads();
  }
}

// ---------------- launcher -------------------------------------------------
extern "C" void kernel_launch(void* const* d_in, const int* in_sizes, int n_in,
                              void* d_out, int out_size, void* d_ws, size_t ws_size,
                              hipStream_t stream) {
  // setup_inputs order: image (unused, shape hardcoded), anchors, deltas, classification
  const float* anchors = (const float*)d_in[1];
  const float* deltas  = (const float*)d_in[2];
  const float* cls     = (const float*)d_in[3];

  float* ws      = (float*)d_ws;
  float* boxes   = ws;                                  // B*N*4      = 392832 f
  float* scoresT = boxes + (size_t)B_ * N_ * 4;         // B*C*N      = 1964160 f
  float* nsc     = scoresT + (size_t)B_ * C_ * N_;      // B*C*300    = 12000 f
  int*   nidx    = (int*)(nsc + (size_t)B_ * C_ * MAXDET); // B*C*300 i32

  rn_decode_kernel<<<(B_ * N_ + 255) / 256, 256, 0, stream>>>(
      anchors, deltas, cls, boxes, scoresT);

  const size_t smem = (size_t)N_ * sizeof(float);       // 196416 B < 320 KB WGP LDS
  (void)hipFuncSetAttribute(reinterpret_cast<const void*>(rn_nms_kernel),
                            hipFuncAttributeMaxDynamicSharedMemorySize, (int)smem);
  rn_nms_kernel<<<B_ * C_, 1024, smem, stream>>>(boxes, scoresT, nsc, nidx);

  rn_topk_kernel<<<B_, 256, 0, stream>>>(boxes, nsc, nidx, (float*)d_out);
}